// dcn_11854109736975
// MI455X (gfx1250) — compile-verified
//
#include <hip/hip_runtime.h>
#include <math.h>

// DCNv2 forward for MI455X (gfx1250, wave32), fp32 on V_WMMA_F32_16X16X4_F32.
// Pipeline: (1) pack weights into WMMA B-fragment order, (2) NCHW->NHWC repack
// of x (makes bilinear gathers b64 instead of 2x b32), (3) fused kernel:
//   phase 1: 27-ch 3x3 conv GEMM (offsets + pre-sigmoid mask), M=32 N=32 K=576
//   phase 2: bilinear sample + main GEMM, M=32 N=64 K=576
// Each wave owns 32 pixels (two 16-pixel M-tiles sharing B fragments).

typedef float v2f __attribute__((ext_vector_type(2)));
typedef float v8f __attribute__((ext_vector_type(8)));

#define Bn  8
#define Cn  64
#define On  64
#define Hn  128
#define Wn  128
#define HWn (Hn*Wn)          // 16384
#define CHWn (Cn*HWn)
#define KTOT 576             // 9 taps * 64 channels
#define KP   (KTOT/2)        // 288 K-pairs

// workspace layout (floats):
//   [0, KP*32*2)             : w27B — B fragments, offset/mask conv (N=32)
//   [.., +KP*64*2)           : wmB  — B fragments, main GEMM (N=64)
//   [.., +B*H*W*C)           : xT   — NHWC copy of x (optional, ws permitting)
#define W27B_OFF 0
#define WMB_OFF  (KP*32*2)
#define XT_OFF   (WMB_OFF + KP*64*2)            // 55296 floats
#define XT_FLOATS ((size_t)Bn*Hn*Wn*Cn)         // 8388608
#define WS_NEED_BYTES ((XT_OFF + XT_FLOATS) * 4ull)

static __device__ __forceinline__ v2f ld2(const float* p) {
    float2 t = *(const float2*)p;
    v2f r; r.x = t.x; r.y = t.y; return r;
}

// ---------------------------------------------------------------------------
// Weight packing: pair j holds K = {2j, 2j+1}; lane n reads pair (k0/2 + hi).
// K = tap*64 + c.  W27[k][n] = n<18 ? w_offset : n<27 ? w_mask : 0.
// ---------------------------------------------------------------------------
__global__ void dcn_pack_weights(const float* __restrict__ w_off,
                                 const float* __restrict__ w_msk,
                                 const float* __restrict__ w_main,
                                 float* __restrict__ ws) {
    int tid = blockIdx.x * blockDim.x + threadIdx.x;
    int j = tid >> 6;
    int n = tid & 63;
    if (j >= KP) return;
    float* w27B = ws + W27B_OFF;
    float* wmB  = ws + WMB_OFF;
#pragma unroll
    for (int e = 0; e < 2; ++e) {
        int k = 2*j + e;
        int t = k >> 6;
        int c = k & 63;
        wmB[(j*64 + n)*2 + e] = w_main[(n*Cn + c)*9 + t];
        if (n < 32) {
            float v = 0.f;
            if (n < 18)      v = w_off[(n*Cn + c)*9 + t];
            else if (n < 27) v = w_msk[((n-18)*Cn + c)*9 + t];
            w27B[(j*32 + n)*2 + e] = v;
        }
    }
}

// ---------------------------------------------------------------------------
// NCHW -> NHWC repack of x via LDS tile transpose (conflict-free, coalesced
// both directions). One block per (b, h, 64-wide w tile).
// ---------------------------------------------------------------------------
__global__ __launch_bounds__(256) void dcn_nchw_to_nhwc(const float* __restrict__ x,
                                                        float* __restrict__ xT) {
    __shared__ float t[64 * 65];
    const int id = blockIdx.x;            // 8*128*2 = 2048 blocks
    const int b  = id >> 8;
    const int rem = id & 255;
    const int h  = rem >> 1;
    const int wt = (rem & 1) * 64;

    const int wlo = threadIdx.x & 63;
    const int qhi = threadIdx.x >> 6;     // 0..3
#pragma unroll 4
    for (int it = 0; it < 16; ++it) {
        const int c = it*4 + qhi;
        t[c*65 + wlo] = x[(((size_t)b*Cn + c)*Hn + h)*Wn + wt + wlo];
    }
    __syncthreads();
#pragma unroll 4
    for (int it = 0; it < 16; ++it) {
        const int ww = it*4 + qhi;
        xT[((((size_t)b*Hn + h)*Wn + wt + ww) << 6) + wlo] = t[wlo*65 + ww];
    }
}

// ---------------------------------------------------------------------------
// Main fused kernel. One wave32 = 32 pixels (row h, columns w0..w0+31) as two
// M-tiles g=0,1 that share every B fragment. 8 waves/block, 512 blocks exact.
// ---------------------------------------------------------------------------
template <bool NHWC>
__global__ __launch_bounds__(256) void dcn_main(const float* __restrict__ x,
                                                const float* __restrict__ xT,
                                                const float* __restrict__ ws,
                                                const float* __restrict__ b_off,
                                                const float* __restrict__ b_msk,
                                                float* __restrict__ out) {
    __shared__ float ylds[8][2*512];   // [wave][g][n][pixel]
    __shared__ float olds[8][1024];    // [wave][o][pixel]  (reused per g)

    const int wave = threadIdx.x >> 5;
    const int lane = threadIdx.x & 31;
    const int m    = lane & 15;
    const int hi   = lane >> 4;

    const int task = blockIdx.x * 8 + wave;        // 0..4095
    const int b    = task >> 9;                    // 512 tasks per image
    const int rem  = task & 511;
    const int h    = rem >> 2;
    const int w0   = (rem & 3) << 5;               // 32-pixel group start

    const float* xb   = x + (size_t)b * CHWn;
    const float* w27B = ws + W27B_OFF;
    const float* wmB  = ws + WMB_OFF;
    float* yl = ylds[wave];
    float* ol = olds[wave];

    // ---------------- Phase 1: 27-channel 3x3 conv --------------------------
    v8f acc[2][2] = {};                            // [g][ntile]
#pragma unroll
    for (int t = 0; t < 9; ++t) {
        const int r  = t / 3, kw = t % 3;
        const int hh = h + r - 1;
        const int hc = min(max(hh, 0), Hn - 1);
        float validg[2]; int ibase[2];
#pragma unroll
        for (int g = 0; g < 2; ++g) {
            const int ww = w0 + 16*g + m + kw - 1;
            validg[g] = (hh >= 0 && hh < Hn && ww >= 0 && ww < Wn) ? 1.f : 0.f;
            const int wc = min(max(ww, 0), Wn - 1);
            ibase[g] = NHWC ? (((b*Hn + hc)*Wn + wc) << 6)   // + c
                            : (hc*Wn + wc);                   // + c*HWn (per-b base in xb)
        }
#pragma unroll 4
        for (int c0 = 0; c0 < 64; c0 += 4) {
            const int cc = c0 + 2*hi;
            const int kp = ((t*64 + c0) >> 1) + hi;
            v2f a[2];
#pragma unroll
            for (int g = 0; g < 2; ++g) {
                if constexpr (NHWC) {
                    a[g] = ld2(&xT[ibase[g] + cc]);
                } else {
                    a[g].x = xb[ibase[g] + (size_t)cc * HWn];
                    a[g].y = xb[ibase[g] + (size_t)(cc+1) * HWn];
                }
                a[g].x *= validg[g];
                a[g].y *= validg[g];
            }
            const v2f bv0 = ld2(&w27B[(kp*32 + m)*2]);
            const v2f bv1 = ld2(&w27B[(kp*32 + 16 + m)*2]);
            acc[0][0] = __builtin_amdgcn_wmma_f32_16x16x4_f32(false, a[0], false, bv0, (short)0, acc[0][0], false, false);
            acc[0][1] = __builtin_amdgcn_wmma_f32_16x16x4_f32(false, a[0], false, bv1, (short)0, acc[0][1], false, false);
            acc[1][0] = __builtin_amdgcn_wmma_f32_16x16x4_f32(false, a[1], false, bv0, (short)0, acc[1][0], false, false);
            acc[1][1] = __builtin_amdgcn_wmma_f32_16x16x4_f32(false, a[1], false, bv1, (short)0, acc[1][1], false, false);
        }
    }

    // bias + redistribute (value (M=pix, N=ch) lives in vgpr jv, lane)
    const float bias0 = b_off[m];
    const int   n1    = 16 + m;
    const float bias1 = (n1 < 18) ? b_off[n1] : (n1 < 27 ? b_msk[n1 - 18] : 0.f);
#pragma unroll
    for (int g = 0; g < 2; ++g)
#pragma unroll
        for (int jv = 0; jv < 8; ++jv) {
            const int pix = jv + 8*hi;
            yl[g*512 + m*16        + pix] = acc[g][0][jv] + bias0;
            yl[g*512 + (16 + m)*16 + pix] = acc[g][1][jv] + bias1;
        }
    __syncthreads();

    // ---------------- Phase 2: deformable sampling + main GEMM --------------
    v8f d[2][4] = {};                              // [g][ntile]
#pragma unroll
    for (int t = 0; t < 9; ++t) {
        float wgt[2][4]; int cof[2][4];
#pragma unroll
        for (int g = 0; g < 2; ++g) {
            const float dy = yl[g*512 + (2*t)*16 + m];
            const float dx = yl[g*512 + (2*t + 1)*16 + m];
            const float mk = 1.f / (1.f + __expf(-yl[g*512 + (18 + t)*16 + m]));
            const float py = dy + (float)(h + t/3 - 1);
            const float px = dx + (float)(w0 + 16*g + m + t%3 - 1);
            const float y0f = floorf(py), x0f = floorf(px);
            const float ly = py - y0f, lx = px - x0f;
            const int y0 = (int)y0f, x0i = (int)x0f;
            const int y1 = y0 + 1,  x1 = x0i + 1;
            const float vy0 = (y0  >= 0 && y0  < Hn) ? 1.f : 0.f;
            const float vy1 = (y1  >= 0 && y1  < Hn) ? 1.f : 0.f;
            const float vx0 = (x0i >= 0 && x0i < Wn) ? 1.f : 0.f;
            const float vx1 = (x1  >= 0 && x1  < Wn) ? 1.f : 0.f;
            const int y0c = min(max(y0, 0), Hn-1), y1c = min(max(y1, 0), Hn-1);
            const int x0c = min(max(x0i,0), Wn-1), x1c = min(max(x1, 0), Wn-1);
            wgt[g][0] = (1.f-ly)*(1.f-lx)*vy0*vx0*mk;
            wgt[g][1] = (1.f-ly)*lx      *vy0*vx1*mk;
            wgt[g][2] = ly      *(1.f-lx)*vy1*vx0*mk;
            wgt[g][3] = ly      *lx      *vy1*vx1*mk;
            if constexpr (NHWC) {
                cof[g][0] = ((b*Hn + y0c)*Wn + x0c) << 6;
                cof[g][1] = ((b*Hn + y0c)*Wn + x1c) << 6;
                cof[g][2] = ((b*Hn + y1c)*Wn + x0c) << 6;
                cof[g][3] = ((b*Hn + y1c)*Wn + x1c) << 6;
            } else {
                cof[g][0] = y0c*Wn + x0c; cof[g][1] = y0c*Wn + x1c;
                cof[g][2] = y1c*Wn + x0c; cof[g][3] = y1c*Wn + x1c;
            }
        }
#pragma unroll 4
        for (int c0 = 0; c0 < 64; c0 += 4) {
            const int cc = c0 + 2*hi;
            const int kp = ((t*64 + c0) >> 1) + hi;
            v2f a[2];
#pragma unroll
            for (int g = 0; g < 2; ++g) {
                if constexpr (NHWC) {
                    // 4 b64 gathers per M-tile (channel pair contiguous in NHWC)
                    const v2f v00 = ld2(&xT[cof[g][0] + cc]);
                    const v2f v01 = ld2(&xT[cof[g][1] + cc]);
                    const v2f v10 = ld2(&xT[cof[g][2] + cc]);
                    const v2f v11 = ld2(&xT[cof[g][3] + cc]);
                    a[g].x = wgt[g][0]*v00.x + wgt[g][1]*v01.x + wgt[g][2]*v10.x + wgt[g][3]*v11.x;
                    a[g].y = wgt[g][0]*v00.y + wgt[g][1]*v01.y + wgt[g][2]*v10.y + wgt[g][3]*v11.y;
                } else {
                    const float* xc = xb + (size_t)cc * HWn;
                    const float* xn = xc + HWn;
                    a[g].x = wgt[g][0]*xc[cof[g][0]] + wgt[g][1]*xc[cof[g][1]]
                           + wgt[g][2]*xc[cof[g][2]] + wgt[g][3]*xc[cof[g][3]];
                    a[g].y = wgt[g][0]*xn[cof[g][0]] + wgt[g][1]*xn[cof[g][1]]
                           + wgt[g][2]*xn[cof[g][2]] + wgt[g][3]*xn[cof[g][3]];
                }
            }
            const v2f b0 = ld2(&wmB[(kp*64 +  0 + m)*2]);
            const v2f b1 = ld2(&wmB[(kp*64 + 16 + m)*2]);
            const v2f b2 = ld2(&wmB[(kp*64 + 32 + m)*2]);
            const v2f b3 = ld2(&wmB[(kp*64 + 48 + m)*2]);
#pragma unroll
            for (int g = 0; g < 2; ++g) {
                d[g][0] = __builtin_amdgcn_wmma_f32_16x16x4_f32(false, a[g], false, b0, (short)0, d[g][0], false, false);
                d[g][1] = __builtin_amdgcn_wmma_f32_16x16x4_f32(false, a[g], false, b1, (short)0, d[g][1], false, false);
                d[g][2] = __builtin_amdgcn_wmma_f32_16x16x4_f32(false, a[g], false, b2, (short)0, d[g][2], false, false);
                d[g][3] = __builtin_amdgcn_wmma_f32_16x16x4_f32(false, a[g], false, b3, (short)0, d[g][3], false, false);
            }
        }
    }

    // ---------------- Store: LDS transpose -> coalesced float4 --------------
#pragma unroll
    for (int g = 0; g < 2; ++g) {
#pragma unroll
        for (int jn = 0; jn < 4; ++jn)
#pragma unroll
            for (int jv = 0; jv < 8; ++jv)
                ol[(jn*16 + m)*16 + jv + 8*hi] = d[g][jn][jv];
        __syncthreads();
        float* ob = out + ((size_t)b * On * HWn) + (size_t)h * Wn + w0 + 16*g;
#pragma unroll
        for (int it = 0; it < 8; ++it) {
            const int f = it*32 + lane;
            const int o = f >> 2;
            const int q = f & 3;
            float4 v = *(const float4*)&ol[o*16 + q*4];
            *(float4*)&ob[(size_t)o * HWn + q*4] = v;
        }
        __syncthreads();
    }
}

// ---------------------------------------------------------------------------
extern "C" void kernel_launch(void* const* d_in, const int* in_sizes, int n_in,
                              void* d_out, int out_size, void* d_ws, size_t ws_size,
                              hipStream_t stream) {
    const float* x      = (const float*)d_in[0];
    const float* w_main = (const float*)d_in[1];
    const float* w_off  = (const float*)d_in[2];
    const float* b_off  = (const float*)d_in[3];
    const float* w_msk  = (const float*)d_in[4];
    const float* b_msk  = (const float*)d_in[5];
    float* out = (float*)d_out;
    float* ws  = (float*)d_ws;

    dcn_pack_weights<<<(KP*64 + 255)/256, 256, 0, stream>>>(w_off, w_msk, w_main, ws);

    const bool use_nhwc = (ws_size >= WS_NEED_BYTES);   // deterministic per-shape
    float* xT = ws + XT_OFF;
    if (use_nhwc) {
        dcn_nchw_to_nhwc<<<2048, 256, 0, stream>>>(x, xT);
        dcn_main<true><<<512, 256, 0, stream>>>(x, xT, ws, b_off, b_msk, out);
    } else {
        dcn_main<false><<<512, 256, 0, stream>>>(x, xT, ws, b_off, b_msk, out);
    }
}